// GNNBackbone_26723286516181
// MI455X (gfx1250) — compile-verified
//
#include <hip/hip_runtime.h>

#define N_NODES 50000
#define N_EDGES 800000
#define NH 8
#define NC 32
#define HCDIM 256
#define EE_MAX (N_EDGES + N_NODES)
#define SLOPE 0.2f
#define EPSV 1e-16f

typedef float v2f __attribute__((ext_vector_type(2)));
typedef float v8f __attribute__((ext_vector_type(8)));

// ---------------------------------------------------------------------------
// Node feature projection: C[N,256] = A[N,K] @ B[K,256] with fp32 WMMA.
// One wave per 16x16 output tile, K stepped by 4 (V_WMMA_F32_16X16X4_F32).
// A-frag layout: lanes 0-15 -> M=0..15 K={k,k+1}; lanes 16-31 -> K={k+2,k+3}.
// B-frag layout: lanes 0-15 -> N=0..15 rows K={k,k+1}; lanes 16-31 -> {k+2,k+3}.
// C/D: VGPR v, lanes 0-15 -> M=v, lanes 16-31 -> M=v+8.
// ---------------------------------------------------------------------------
template <int K>
__global__ __launch_bounds__(256) void gat_gemm_wmma(
    const float* __restrict__ A, const float* __restrict__ B,
    float* __restrict__ C) {
  const int lane = threadIdx.x & 31;
  const int wv   = threadIdx.x >> 5;
  const int tile = blockIdx.x * 8 + wv;   // (N/16)*16 = 50000 tiles exactly
  const int mt   = tile >> 4;
  const int nt   = tile & 15;
  const int half = lane >> 4;
  const int l15  = lane & 15;

  const float* Arow = A + (size_t)(mt * 16 + l15) * K;
  const float* Bcol = B + nt * 16 + l15;

  v8f acc = {0.f, 0.f, 0.f, 0.f, 0.f, 0.f, 0.f, 0.f};
#pragma unroll
  for (int k = 0; k < K; k += 4) {
    const int kb = k + half * 2;
    v2f a, b;
    a.x = Arow[kb];
    a.y = Arow[kb + 1];
    b.x = Bcol[(size_t)kb * HCDIM];
    b.y = Bcol[(size_t)(kb + 1) * HCDIM];
    acc = __builtin_amdgcn_wmma_f32_16x16x4_f32(false, a, false, b,
                                                (short)0, acc, false, false);
  }
  const int ncol = nt * 16 + l15;
  const int mrow = mt * 16 + half * 8;
#pragma unroll
  for (int v = 0; v < 8; ++v)
    C[(size_t)(mrow + v) * HCDIM + ncol] = acc[v];
}

// ---------------------------------------------------------------------------
// Per-node attention coefficients: alS[n,h] = sum_c h[n,h,c]*a_s[h,c] (and alD)
// Block = 1 node, 8 waves = 8 heads, wave32 shuffle reduction.
// ---------------------------------------------------------------------------
__global__ __launch_bounds__(256) void gat_alphas(
    const float* __restrict__ h, const float* __restrict__ as,
    const float* __restrict__ ad, float* __restrict__ alS,
    float* __restrict__ alD) {
  const int n  = blockIdx.x;
  const int hd = threadIdx.x >> 5;
  const int c  = threadIdx.x & 31;
  const float v = h[(size_t)n * HCDIM + hd * NC + c];
  float s = v * as[hd * NC + c];
  float d = v * ad[hd * NC + c];
#pragma unroll
  for (int off = 16; off > 0; off >>= 1) {
    s += __shfl_xor(s, off, 32);
    d += __shfl_xor(d, off, 32);
  }
  if (c == 0) {
    alS[n * NH + hd] = s;
    alD[n * NH + hd] = d;
  }
}

// w_e[h] = sum_c We[h,c]*ae[h,c]   (tiny: 1 block)
__global__ void gat_we(const float* __restrict__ We,
                       const float* __restrict__ ae, float* __restrict__ wE) {
  const int hd = threadIdx.x >> 5;
  const int c  = threadIdx.x & 31;
  float v = We[hd * NC + c] * ae[hd * NC + c];
#pragma unroll
  for (int off = 16; off > 0; off >>= 1) v += __shfl_xor(v, off, 32);
  if (c == 0) wE[hd] = v;
}

// out = bias (pre-aggregation init), m-keys = 0 (== -NaN sentinel), denom = 0
__global__ void gat_init(float* __restrict__ out, const float* __restrict__ bias,
                         unsigned* __restrict__ mU, float* __restrict__ denom) {
  const int i = blockIdx.x * blockDim.x + threadIdx.x;  // N*256 exact
  out[i] = bias[i & (HCDIM - 1)];
  if (i < N_NODES * NH) {
    mU[i]    = 0u;
    denom[i] = 0.f;
  }
}

__global__ void gat_zero_ls(float* cnt, float* sea) {
  const int i = blockIdx.x * blockDim.x + threadIdx.x;
  if (i < N_NODES) { cnt[i] = 0.f; sea[i] = 0.f; }
}

// incoming-edge count + edge_attr sum per node (self-loop fill_value='mean')
__global__ void gat_loop_stats(const int* __restrict__ dst,
                               const float* __restrict__ ea, float* cnt,
                               float* sea) {
  const int i = blockIdx.x * blockDim.x + threadIdx.x;
  if (i < N_EDGES) {
    atomicAdd(&cnt[dst[i]], 1.f);
    atomicAdd(&sea[dst[i]], ea[i]);
  }
}

// monotone float <-> uint key mapping (so u32 atomicMax == float max)
__device__ __forceinline__ unsigned flipf(float f) {
  unsigned u = __float_as_uint(f);
  return u ^ ((u >> 31) ? 0xFFFFFFFFu : 0x80000000u);
}
__device__ __forceinline__ float unflipf(unsigned u) {
  return __uint_as_float(u ^ ((u >> 31) ? 0x80000000u : 0xFFFFFFFFu));
}

__device__ __forceinline__ float edge_logit(
    int e, int hd, const int* __restrict__ src, const int* __restrict__ dst,
    const float* __restrict__ ea, const float* __restrict__ alS,
    const float* __restrict__ alD, const float* __restrict__ wE,
    const float* __restrict__ cnt, const float* __restrict__ sea, int& s,
    int& d) {
  float a;
  if (e < N_EDGES) {
    s = src[e]; d = dst[e]; a = ea[e];
  } else {                       // self loop with mean incoming edge_attr
    s = d = e - N_EDGES;
    a = sea[s] / fmaxf(cnt[s], 1.f);
  }
  float lg = alS[s * NH + hd] + alD[d * NH + hd] + a * wE[hd];
  return lg > 0.f ? lg : SLOPE * lg;   // LeakyReLU(0.2)
}

// segment max of logits over dst
__global__ void gat_segmax(const int* __restrict__ src,
                           const int* __restrict__ dst,
                           const float* __restrict__ ea,
                           const float* __restrict__ alS,
                           const float* __restrict__ alD,
                           const float* __restrict__ wE,
                           const float* __restrict__ cnt,
                           const float* __restrict__ sea,
                           unsigned* __restrict__ mU, int nEdges) {
  const int i = blockIdx.x * blockDim.x + threadIdx.x;
  if (i >= nEdges * NH) return;
  int s, d;
  const float lg = edge_logit(i >> 3, i & 7, src, dst, ea, alS, alD, wE, cnt, sea, s, d);
  atomicMax(&mU[d * NH + (i & 7)], flipf(lg));
}

// denom[dst,h] = sum exp(logit - m[dst,h])
__global__ void gat_denom(const int* __restrict__ src,
                          const int* __restrict__ dst,
                          const float* __restrict__ ea,
                          const float* __restrict__ alS,
                          const float* __restrict__ alD,
                          const float* __restrict__ wE,
                          const float* __restrict__ cnt,
                          const float* __restrict__ sea,
                          const unsigned* __restrict__ mU,
                          float* __restrict__ denom, int nEdges) {
  const int i = blockIdx.x * blockDim.x + threadIdx.x;
  if (i >= nEdges * NH) return;
  int s, d;
  const int hd = i & 7;
  const float lg = edge_logit(i >> 3, hd, src, dst, ea, alS, alD, wE, cnt, sea, s, d);
  atomicAdd(&denom[d * NH + hd], expf(lg - unflipf(mU[d * NH + hd])));
}

// out[dst] += h[src] * alpha  — one wave per edge, lane = channel, loop heads.
// alpha recomputed per edge (8 logit+exp) instead of storing a 27MB buffer.
__global__ __launch_bounds__(256) void gat_aggregate(
    const float* __restrict__ hfeat, const int* __restrict__ src,
    const int* __restrict__ dst, const float* __restrict__ ea,
    const float* __restrict__ alS, const float* __restrict__ alD,
    const float* __restrict__ wE, const float* __restrict__ cnt,
    const float* __restrict__ sea, const unsigned* __restrict__ mU,
    const float* __restrict__ denom, float* __restrict__ out, int nEdges) {
  const int e = blockIdx.x * 8 + (threadIdx.x >> 5);
  if (e >= nEdges) return;
  const int lane = threadIdx.x & 31;
  int s, d;
  float alpha = 0.f;
  if (lane < NH) {
    const float lg = edge_logit(e, lane, src, dst, ea, alS, alD, wE, cnt, sea, s, d);
    const float pv = expf(lg - unflipf(mU[d * NH + lane]));
    alpha = pv / (denom[d * NH + lane] + EPSV);
  }
  if (e < N_EDGES) { s = src[e]; d = dst[e]; } else { s = d = e - N_EDGES; }
  const float* hs = hfeat + (size_t)s * HCDIM + lane;
  float* od       = out + (size_t)d * HCDIM + lane;
#pragma unroll
  for (int hd = 0; hd < NH; ++hd) {
    const float av = __shfl(alpha, hd, 32);
    atomicAdd(od + hd * NC, hs[hd * NC] * av);
  }
}

__global__ void gat_relu(float* __restrict__ x, int n) {
  const int i = blockIdx.x * blockDim.x + threadIdx.x;
  if (i < n) x[i] = fmaxf(x[i], 0.f);
}

// ---------------------------------------------------------------------------
extern "C" void kernel_launch(void* const* d_in, const int* in_sizes, int n_in,
                              void* d_out, int out_size, void* d_ws,
                              size_t ws_size, hipStream_t stream) {
  const float* x  = (const float*)d_in[0];
  const int*   ei = (const int*)d_in[1];
  const float* ea = (const float*)d_in[2];
  const int* src = ei;
  const int* dst = ei + N_EDGES;

  const float* W[3]  = {(const float*)d_in[3], (const float*)d_in[9],  (const float*)d_in[15]};
  const float* As[3] = {(const float*)d_in[4], (const float*)d_in[10], (const float*)d_in[16]};
  const float* Ad[3] = {(const float*)d_in[5], (const float*)d_in[11], (const float*)d_in[17]};
  const float* We[3] = {(const float*)d_in[6], (const float*)d_in[12], (const float*)d_in[18]};
  const float* Ae[3] = {(const float*)d_in[7], (const float*)d_in[13], (const float*)d_in[19]};
  const float* Bb[3] = {(const float*)d_in[8], (const float*)d_in[14], (const float*)d_in[20]};

  // workspace layout (floats)
  float* ws = (float*)d_ws;
  size_t off = 0;
  float*    hA    = ws + off; off += (size_t)N_NODES * HCDIM;  // projected feats
  float*    alS   = ws + off; off += (size_t)N_NODES * NH;
  float*    alD   = ws + off; off += (size_t)N_NODES * NH;
  unsigned* mU    = (unsigned*)(ws + off); off += (size_t)N_NODES * NH;
  float*    denom = ws + off; off += (size_t)N_NODES * NH;
  float*    cnt   = ws + off; off += N_NODES;
  float*    sea   = ws + off; off += N_NODES;
  float*    wE    = ws + off; off += 8;

  float* hOut = (float*)d_out;  // layer output ping buffer; final layer lands here

  // self-loop edge_attr stats (shared by layers 2 and 3)
  gat_zero_ls<<<(N_NODES + 255) / 256, 256, 0, stream>>>(cnt, sea);
  gat_loop_stats<<<(N_EDGES + 255) / 256, 256, 0, stream>>>(dst, ea, cnt, sea);

  const float* hin = x;
  for (int l = 0; l < 3; ++l) {
    const int nE = (l == 0) ? N_EDGES : EE_MAX;  // layer 1: no self loops
    const int th = nE * NH;

    if (l == 0)
      gat_gemm_wmma<4><<<50000 / 8, 256, 0, stream>>>(hin, W[l], hA);
    else
      gat_gemm_wmma<HCDIM><<<50000 / 8, 256, 0, stream>>>(hin, W[l], hA);

    gat_alphas<<<N_NODES, 256, 0, stream>>>(hA, As[l], Ad[l], alS, alD);
    gat_we<<<1, 256, 0, stream>>>(We[l], Ae[l], wE);
    gat_init<<<(N_NODES * HCDIM) / 256, 256, 0, stream>>>(hOut, Bb[l], mU, denom);
    gat_segmax<<<(th + 255) / 256, 256, 0, stream>>>(src, dst, ea, alS, alD, wE,
                                                     cnt, sea, mU, nE);
    gat_denom<<<(th + 255) / 256, 256, 0, stream>>>(src, dst, ea, alS, alD, wE,
                                                    cnt, sea, mU, denom, nE);
    gat_aggregate<<<(nE + 7) / 8, 256, 0, stream>>>(hA, src, dst, ea, alS, alD,
                                                    wE, cnt, sea, mU, denom,
                                                    hOut, nE);
    gat_relu<<<(N_NODES * HCDIM) / 256, 256, 0, stream>>>(hOut, N_NODES * HCDIM);
    hin = hOut;
  }
}